// GCN_41094247088649
// MI455X (gfx1250) — compile-verified
//
#include <hip/hip_runtime.h>

#define N_NODES 100000
#define N_EDGES 800000
#define N_GRAPHS 64
#define IN_DIM 200
#define HID 128
#define ROWS_PB 32                       // rows per block in GEMM1 (100000/32 = 3125)

typedef float v2f __attribute__((ext_vector_type(2)));
typedef float v8f __attribute__((ext_vector_type(8)));
typedef unsigned int u32x4 __attribute__((ext_vector_type(4)));
typedef int i32x4 __attribute__((ext_vector_type(4)));
typedef int i32x8 __attribute__((ext_vector_type(8)));

// ---------------- degree / norm ----------------
__global__ void k_deg(const int* __restrict__ dst, const float* __restrict__ ew,
                      float* __restrict__ deg, int nE) {
  int e = blockIdx.x * blockDim.x + threadIdx.x;
  if (e < nE) atomicAdd(&deg[dst[e]], ew[e]);
}

__global__ void k_dis(float* __restrict__ deg, int n) {
  int i = blockIdx.x * blockDim.x + threadIdx.x;
  if (i < n) deg[i] = rsqrtf(deg[i] + 1.0f);   // deg -> dis, in place
}

__global__ void k_norm(const int* __restrict__ src, const int* __restrict__ dst,
                       const float* __restrict__ ew, const float* __restrict__ dis,
                       float* __restrict__ norm, int nE) {
  int e = blockIdx.x * blockDim.x + threadIdx.x;
  if (e < nE) norm[e] = dis[src[e]] * ew[e] * dis[dst[e]];
}

// ---------------- GEMM1: xw1 = x @ W1 via fp32 WMMA ----------------
// Block: 256 threads = 8 waves. Block computes rows [32*bx, 32*bx+32) x all 128 cols.
// Wave w owns the 16-col slice [16w, 16w+16) and keeps TWO 16x16 accumulators
// (row halves 0-15 and 16-31) that share each B fragment -> half the W1 traffic
// and two independent WMMAs per K-step (no D->A/B RAW hazard between them).
// The 32x200 x-tile is DMA'd into LDS by the Tensor Data Mover (D# per ISA ch.8).
__global__ __launch_bounds__(256) void k_gemm1(const float* __restrict__ x,
                                               const float* __restrict__ W1,
                                               float* __restrict__ xw1) {
  __shared__ float xs[ROWS_PB][IN_DIM];                  // 25.6 KB
  const int block_row = blockIdx.x * ROWS_PB;
  const int tid = threadIdx.x;

#if defined(__AMDGCN__) && __has_builtin(__builtin_amdgcn_tensor_load_to_lds)
  if (tid < 32) {                                        // wave 0 issues the TDM copy
    unsigned lds_base = (unsigned)(unsigned long long)(&xs[0][0]);
    unsigned long long ga =
        (unsigned long long)(const void*)(x + (unsigned long long)block_row * IN_DIM);
    // D# group 0: count=1 (valid user desc), lds_addr, global_addr[56:0], type=2
    u32x4 g0 = { 1u,
                 lds_base,
                 (unsigned)(ga & 0xFFFFFFFFu),
                 (unsigned)((ga >> 32) & 0x01FFFFFFu) | (2u << 30) };
    // D# group 1: data_size=4B(code 2); tensor_dim0=200; tensor_dim1=100000;
    //             tile_dim0=200; tile_dim1=32; tensor_dim0_stride=200
    i32x8 g1 = { (int)(2u << 16),                                   // data_size
                 (int)((unsigned)IN_DIM << 16),                     // dim0[15:0] @ [31:16]
                 (int)(((unsigned)N_NODES & 0xFFFFu) << 16),        // dim0 hi=0 | dim1 lo
                 (int)(((unsigned)N_NODES >> 16) |
                       ((unsigned)IN_DIM << 16)),                   // dim1 hi | tile_dim0
                 (int)ROWS_PB,                                      // tile_dim1 | tile_dim2=0
                 (int)IN_DIM,                                       // dim0_stride lo32
                 0, 0 };
    i32x4 gz = { 0, 0, 0, 0 };
#if __clang_major__ >= 23
    i32x8 gz8 = { 0, 0, 0, 0, 0, 0, 0, 0 };
    __builtin_amdgcn_tensor_load_to_lds(g0, g1, gz, gz, gz8, 0);
#else
    __builtin_amdgcn_tensor_load_to_lds(g0, g1, gz, gz, 0);
#endif
    __builtin_amdgcn_s_wait_tensorcnt(0);
  }
#else
  for (int i = tid; i < ROWS_PB * IN_DIM; i += 256) {
    int r = i / IN_DIM, c = i % IN_DIM;
    xs[r][c] = x[(size_t)(block_row + r) * IN_DIM + c];
  }
#endif
  __syncthreads();

  const int wave = tid >> 5;
  const int lane = tid & 31;
  const int col0 = wave * 16;
  const int m    = lane & 15;          // A row-in-half-tile; B/D column-in-slice
  const int kh   = (lane >> 4) * 2;    // lanes 0-15: K=k,k+1 ; lanes 16-31: K=k+2,k+3

  v8f acc0 = {};                       // rows block_row + 0..15
  v8f acc1 = {};                       // rows block_row + 16..31
  for (int k = 0; k < IN_DIM; k += 4) {
    __builtin_prefetch(&W1[(size_t)(k + 8) * HID + col0 + m], 0, 3);
    v2f b;                                   // B: 4x16 f32 fragment (shared by both tiles)
    b.x = W1[(size_t)(k + kh)     * HID + col0 + m];
    b.y = W1[(size_t)(k + kh + 1) * HID + col0 + m];
    v2f a0, a1;                              // A: 16x4 f32 fragments
    a0.x = xs[m][k + kh];
    a0.y = xs[m][k + kh + 1];
    a1.x = xs[16 + m][k + kh];
    a1.y = xs[16 + m][k + kh + 1];
    acc0 = __builtin_amdgcn_wmma_f32_16x16x4_f32(false, a0, false, b,
                                                 (short)0, acc0, false, false);
    acc1 = __builtin_amdgcn_wmma_f32_16x16x4_f32(false, a1, false, b,
                                                 (short)0, acc1, false, false);
  }

  // D layout: VGPR i -> row i (lanes 0-15) / row i+8 (lanes 16-31), col = lane&15
#pragma unroll
  for (int i = 0; i < 8; ++i) {
    int r0 = block_row + i + ((lane >> 4) * 8);
    xw1[(size_t)r0 * HID + col0 + m]        = acc0[i];
    xw1[(size_t)(r0 + 16) * HID + col0 + m] = acc1[i];
  }
}

// ---------------- edge scatter (feature-parallel) ----------------
__global__ void k_scatter1(const int* __restrict__ src, const int* __restrict__ dst,
                           const float* __restrict__ norm, const float* __restrict__ xw1,
                           float* __restrict__ agg1) {
  size_t gid = (size_t)blockIdx.x * blockDim.x + threadIdx.x;   // e*128 + f
  int e = (int)(gid >> 7);
  int f = (int)(gid & 127);
  if (e < N_EDGES) {
    int s = src[e], d = dst[e];
    atomicAdd(&agg1[(size_t)d * HID + f], xw1[(size_t)s * HID + f] * norm[e]);
  }
}

// ---------------- fused bias + self-loop + relu: h (in place over agg1) ----------------
__global__ void k_h(float* __restrict__ agg1, const float* __restrict__ xw1,
                    const float* __restrict__ dis, const float* __restrict__ b1) {
  size_t gid = (size_t)blockIdx.x * blockDim.x + threadIdx.x;   // i*128 + f
  int i = (int)(gid >> 7);
  int f = (int)(gid & 127);
  if (i < N_NODES) {
    float d = dis[i];
    float v = agg1[gid] + xw1[gid] * (d * d) + b1[f];
    agg1[gid] = fmaxf(v, 0.0f);
  }
}

// ---------------- GEMM2: 128-wide matvec ----------------
__global__ void k_gemm2(const float* __restrict__ h, const float* __restrict__ W2,
                        float* __restrict__ xw2) {
  int i = blockIdx.x * blockDim.x + threadIdx.x;
  if (i >= N_NODES) return;
  const float* hp = h + (size_t)i * HID;
  float s = 0.0f;
#pragma unroll 8
  for (int f = 0; f < HID; ++f) s = fmaf(hp[f], W2[f], s);
  xw2[i] = s;
}

__global__ void k_scatter2(const int* __restrict__ src, const int* __restrict__ dst,
                           const float* __restrict__ norm, const float* __restrict__ xw2,
                           float* __restrict__ agg2) {
  int e = blockIdx.x * blockDim.x + threadIdx.x;
  if (e < N_EDGES) atomicAdd(&agg2[dst[e]], xw2[src[e]] * norm[e]);
}

__global__ void k_final(const float* __restrict__ agg2, const float* __restrict__ xw2,
                        const float* __restrict__ dis, const float* __restrict__ b2,
                        const int* __restrict__ batch,
                        float* __restrict__ gsum, float* __restrict__ gcnt) {
  int i = blockIdx.x * blockDim.x + threadIdx.x;
  if (i >= N_NODES) return;
  float d = dis[i];
  float v = agg2[i] + xw2[i] * (d * d) + b2[0];
  int g = batch[i];
  atomicAdd(&gsum[g], v);
  atomicAdd(&gcnt[g], 1.0f);
}

__global__ void k_out(const float* __restrict__ gsum, const float* __restrict__ gcnt,
                      float* __restrict__ out) {
  int g = threadIdx.x;
  if (g < N_GRAPHS) out[g] = gsum[g] / fmaxf(gcnt[g], 1.0f);
}

// ---------------- launcher ----------------
extern "C" void kernel_launch(void* const* d_in, const int* in_sizes, int n_in,
                              void* d_out, int out_size, void* d_ws, size_t ws_size,
                              hipStream_t stream) {
  const float* x   = (const float*)d_in[0];
  const int*   ei  = (const int*)d_in[1];     // [2, E]
  const float* ew  = (const float*)d_in[2];
  const int*   bat = (const int*)d_in[3];
  const float* W1  = (const float*)d_in[4];
  const float* b1  = (const float*)d_in[5];
  const float* W2  = (const float*)d_in[6];
  const float* b2  = (const float*)d_in[7];
  float* out = (float*)d_out;

  // workspace layout (floats); zeroed regions are contiguous at the front
  float* ws   = (float*)d_ws;
  float* deg  = ws;                                  // N      (becomes dis)
  float* agg2 = deg  + N_NODES;                      // N
  float* gsum = agg2 + N_NODES;                      // 64
  float* gcnt = gsum + N_GRAPHS;                     // 64
  float* agg1 = gcnt + N_GRAPHS;                     // N*128  (becomes h)
  float* norm = agg1 + (size_t)N_NODES * HID;        // E
  float* xw1  = norm + N_EDGES;                      // N*128
  float* xw2  = xw1  + (size_t)N_NODES * HID;        // N

  size_t zero_bytes = ((size_t)2 * N_NODES + 2 * N_GRAPHS + (size_t)N_NODES * HID)
                      * sizeof(float);
  hipMemsetAsync(deg, 0, zero_bytes, stream);

  const int* src = ei;
  const int* dst = ei + N_EDGES;

  k_deg <<<(N_EDGES + 255) / 256, 256, 0, stream>>>(dst, ew, deg, N_EDGES);
  k_dis <<<(N_NODES + 255) / 256, 256, 0, stream>>>(deg, N_NODES);
  k_norm<<<(N_EDGES + 255) / 256, 256, 0, stream>>>(src, dst, ew, deg, norm, N_EDGES);

  k_gemm1<<<N_NODES / ROWS_PB, 256, 0, stream>>>(x, W1, xw1);            // 3125 blocks

  k_scatter1<<<(int)(((size_t)N_EDGES * HID) / 256), 256, 0, stream>>>(src, dst, norm, xw1, agg1);
  k_h       <<<(int)(((size_t)N_NODES * HID) / 256), 256, 0, stream>>>(agg1, xw1, deg, b1);

  k_gemm2   <<<(N_NODES + 255) / 256, 256, 0, stream>>>(agg1, W2, xw2);
  k_scatter2<<<(N_EDGES + 255) / 256, 256, 0, stream>>>(src, dst, norm, xw2, agg2);
  k_final   <<<(N_NODES + 255) / 256, 256, 0, stream>>>(agg2, xw2, deg, b2, bat, gsum, gcnt);
  k_out     <<<1, 64, 0, stream>>>(gsum, gcnt, out);
}